// ExponentialMovingAverage_75033078661344
// MI455X (gfx1250) — compile-verified
//
#include <hip/hip_runtime.h>

// EMA scan as blocked lower-triangular WMMA:
//   per 16-step tile:  Y = A * X + carry,  A[m][k] = w*(1-w)^(m-k) (k<=m)
//   carry term:        Y[m][n] += (1-w)^(m+1) * acc_in[n]
//   new carry:         acc = Y[15][:]  (D VGPR7, lanes 16-31) -> ds_bpermute
//
// Memory-bound: ~1.05 GB total traffic -> ~45us floor at 23.3 TB/s.
// Per-lane loads/stores are contiguous along T (the fast axis), so every
// cacheline is fully consumed; 4096 waves cover the device.

typedef float v2f __attribute__((ext_vector_type(2)));
typedef float v4f __attribute__((ext_vector_type(4)));
typedef float v8f __attribute__((ext_vector_type(8)));

#define B_ 8
#define C_ 64
#define F_ 128
#define T_ 2000
#define TILE 16
#define NTILES (T_ / TILE)   // 125
#define FB (F_ / 16)         // 8 f-blocks per (b,c)

__device__ __forceinline__ float pow_small(float a, int e) {
    // a^e for 0 <= e <= 16, exact-enough repeated multiply (runs once per wave)
    float r = 1.0f;
    for (int i = 0; i < e; ++i) r *= a;
    return r;
}

__global__ __launch_bounds__(256)
void ema_wmma_kernel(const float* __restrict__ x,      // [B,C,F,T]
                     const float* __restrict__ init,   // [B,C,F,1]
                     const float* __restrict__ wts,    // [C]
                     float* __restrict__ y,            // [B,C,F,T]
                     float* __restrict__ finals) {     // [B,C,F]
    const int lane = threadIdx.x & 31;
    const int wave = blockIdx.x * 8 + (threadIdx.x >> 5);  // 0..4095
    const int fblk = wave & (FB - 1);
    const int bc   = wave / FB;            // 0..511 (b*C + c)
    const int c    = bc & (C_ - 1);
    const int n    = lane & 15;            // N column (f within block) / A-matrix row M
    const int half = lane >> 4;

    float w = wts[c];
    w = fminf(fmaxf(w, 0.0f), 1.0f);
    const float a = 1.0f - w;

    // ---- A-matrix operands, f32 16x16x4 layout:
    // lanes 0-15:  M=lane,    VGPR0=K(kb+0), VGPR1=K(kb+1)
    // lanes 16-31: M=lane-16, VGPR0=K(kb+2), VGPR1=K(kb+3)
    const int m  = n;
    const int k0 = 2 * half;
    v2f A0, A1, A2, A3;
    {
        auto amk = [&](int k) -> float {
            return (k <= m) ? w * pow_small(a, m - k) : 0.0f;
        };
        A0.x = amk(0  + k0); A0.y = amk(0  + k0 + 1);
        A1.x = amk(4  + k0); A1.y = amk(4  + k0 + 1);
        A2.x = amk(8  + k0); A2.y = amk(8  + k0 + 1);
        A3.x = amk(12 + k0); A3.y = amk(12 + k0 + 1);
    }

    // Carry powers: D VGPR r holds row M = r + 8*half -> needs (1-w)^(M+1)
    v8f p8;
#pragma unroll
    for (int r = 0; r < 8; ++r) p8[r] = pow_small(a, r + 8 * half + 1);

    // Per-lane row pointers (T is the contiguous axis)
    const size_t row = (size_t)(bc * F_ + fblk * 16 + n);
    const float* __restrict__ xrow = x + row * T_;
    float* __restrict__       yrow = y + row * T_;

    // Initial carry: acc[n] (lanes n and n+16 read the same value)
    float accv = init[row];

    for (int tile = 0; tile < NTILES; ++tile) {
        const int t0 = tile * TILE;

        // B operand: lane holds X[t0+4j+2*half .. +1][column n] -> direct float2 load
        v2f b0 = *(const v2f*)(xrow + t0 + 0  + k0);
        v2f b1 = *(const v2f*)(xrow + t0 + 4  + k0);
        v2f b2 = *(const v2f*)(xrow + t0 + 8  + k0);
        v2f b3 = *(const v2f*)(xrow + t0 + 12 + k0);

        if (tile + 4 < NTILES)
            __builtin_prefetch(xrow + t0 + 4 * TILE, 0, 1);

        // Seed accumulator with carry contribution: D[r] = (1-w)^(M+1) * acc_in[n]
        v8f d;
#pragma unroll
        for (int r = 0; r < 8; ++r) d[r] = p8[r] * accv;

        // Chain K-accumulation: D += A_j x B_j  (K = 4j..4j+3)
        d = __builtin_amdgcn_wmma_f32_16x16x4_f32(false, A0, false, b0, (short)0, d, false, false);
        d = __builtin_amdgcn_wmma_f32_16x16x4_f32(false, A1, false, b1, (short)0, d, false, false);
        d = __builtin_amdgcn_wmma_f32_16x16x4_f32(false, A2, false, b2, (short)0, d, false, false);
        d = __builtin_amdgcn_wmma_f32_16x16x4_f32(false, A3, false, b3, (short)0, d, false, false);

        // Store: lane owns t = t0 + 8*half + {0..7} -> two contiguous b128 stores
        v4f lo = { d[0], d[1], d[2], d[3] };
        v4f hi = { d[4], d[5], d[6], d[7] };
        *(v4f*)(yrow + t0 + 8 * half + 0) = lo;
        *(v4f*)(yrow + t0 + 8 * half + 4) = hi;

        // New carry = Y[15][n] = d[7] on lane 16+n; broadcast to both halves
        accv = __int_as_float(
            __builtin_amdgcn_ds_bpermute((16 + n) * 4, __float_as_int(d[7])));
    }

    // Final state output [B,C,F,1]
    if (half == 0) finals[row] = accv;
}

extern "C" void kernel_launch(void* const* d_in, const int* in_sizes, int n_in,
                              void* d_out, int out_size, void* d_ws, size_t ws_size,
                              hipStream_t stream) {
    const float* x    = (const float*)d_in[0];   // mag_spec [B,C,F,T]
    const float* init = (const float*)d_in[1];   // initial_state [B,C,F,1]
    const float* wts  = (const float*)d_in[2];   // weights [C]

    float* y      = (float*)d_out;                        // [B,C,F,T]
    float* finals = y + (size_t)B_ * C_ * F_ * T_;        // [B,C,F,1]

    // 4096 waves total: one wave per (b,c, 16-wide f block); full waves only.
    dim3 grid((B_ * C_ * FB) / 8);   // 512 blocks
    dim3 block(256);                 // 8 waves/block
    ema_wmma_kernel<<<grid, block, 0, stream>>>(x, init, wts, y, finals);
}